// rf_44186623541738
// MI455X (gfx1250) — compile-verified
//
#include <hip/hip_runtime.h>
#include <cfloat>
#include <cmath>

typedef float v2f __attribute__((ext_vector_type(2)));
typedef float v8f __attribute__((ext_vector_type(8)));

#define B_ 64
#define S_ 128
#define F_ 136
#define H_ 128

// ---------------------------------------------------------------------------
// Kernel A: fused MLP with fp32 WMMA (V_WMMA_F32_16X16X4_F32).
//   reprs = relu(x @ W1 + b1) @ W2 + b2            [8192, 128]
//   pre   = reprs @ com_W1 + com_b1                [8192, 128]  (distributed bias trick)
// 512 blocks x 256 threads (8 waves). Each block: 16 rows; each wave: 16 cols.
// ---------------------------------------------------------------------------
__global__ __launch_bounds__(256) void fused_mlp_kernel(
    const float* __restrict__ x,
    const float* __restrict__ W1, const float* __restrict__ b1,
    const float* __restrict__ W2, const float* __restrict__ b2,
    const float* __restrict__ cW1, const float* __restrict__ cb1,
    float* __restrict__ reprs_ws, float* __restrict__ pre_ws)
{
    __shared__ float xs[16 * 137];   // 16 x 136 input tile, padded (gcd(137,64)=1)
    __shared__ float hs[16 * 129];   // 16 x 128 hidden tile, padded
    __shared__ float rs[16 * 129];   // 16 x 128 reprs tile, padded

    const int t  = threadIdx.x;
    const int r0 = blockIdx.x * 16;

    for (int i = t; i < 16 * F_; i += 256) {
        int m = i / F_, k = i % F_;
        xs[m * 137 + k] = x[(r0 + m) * F_ + k];
    }
    __syncthreads();

    const int lane = t & 31;
    const int wid  = t >> 5;
    const int n0   = wid * 16;       // this wave's output-column tile
    const int half = lane >> 4;      // 0: K pair {k0,k0+1}, 1: {k0+2,k0+3}
    const int nl   = lane & 15;      // row (A) / col (B,C) within tile

    // ---- GEMM1: hidden = relu(x @ W1 + b1), K = 136 ----
    v8f c = {};
    for (int k0 = 0; k0 < F_; k0 += 4) {
        v2f a, bm;
        a.x  = xs[nl * 137 + k0 + 2 * half];
        a.y  = xs[nl * 137 + k0 + 2 * half + 1];
        bm.x = W1[(k0 + 2 * half    ) * H_ + n0 + nl];
        bm.y = W1[(k0 + 2 * half + 1) * H_ + n0 + nl];
        c = __builtin_amdgcn_wmma_f32_16x16x4_f32(false, a, false, bm,
                                                  (short)0, c, false, false);
    }
    {
        float bias = b1[n0 + nl];
        #pragma unroll
        for (int i = 0; i < 8; ++i) {
            float v = c[i] + bias;                    // (m = i + 8*half, n = n0+nl)
            hs[(i + 8 * half) * 129 + n0 + nl] = v > 0.f ? v : 0.f;
        }
    }
    __syncthreads();

    // ---- GEMM2: reprs = hidden @ W2 + b2, K = 128 ----
    v8f c2 = {};
    for (int k0 = 0; k0 < H_; k0 += 4) {
        v2f a, bm;
        a.x  = hs[nl * 129 + k0 + 2 * half];
        a.y  = hs[nl * 129 + k0 + 2 * half + 1];
        bm.x = W2[(k0 + 2 * half    ) * H_ + n0 + nl];
        bm.y = W2[(k0 + 2 * half + 1) * H_ + n0 + nl];
        c2 = __builtin_amdgcn_wmma_f32_16x16x4_f32(false, a, false, bm,
                                                   (short)0, c2, false, false);
    }
    {
        float bias = b2[n0 + nl];
        #pragma unroll
        for (int i = 0; i < 8; ++i) {
            float v = c2[i] + bias;
            int   m = i + 8 * half;
            rs[m * 129 + n0 + nl] = v;
            reprs_ws[(r0 + m) * H_ + n0 + nl] = v;
        }
    }
    __syncthreads();

    // ---- GEMM3: pre = reprs @ com_W1 + com_b1, K = 128 ----
    v8f c3 = {};
    for (int k0 = 0; k0 < H_; k0 += 4) {
        v2f a, bm;
        a.x  = rs[nl * 129 + k0 + 2 * half];
        a.y  = rs[nl * 129 + k0 + 2 * half + 1];
        bm.x = cW1[(k0 + 2 * half    ) * H_ + n0 + nl];
        bm.y = cW1[(k0 + 2 * half + 1) * H_ + n0 + nl];
        c3 = __builtin_amdgcn_wmma_f32_16x16x4_f32(false, a, false, bm,
                                                   (short)0, c3, false, false);
    }
    {
        float bias = cb1[n0 + nl];
        #pragma unroll
        for (int i = 0; i < 8; ++i) {
            int m = i + 8 * half;
            pre_ws[(r0 + m) * H_ + n0 + nl] = c3[i] + bias;
        }
    }
}

// ---------------------------------------------------------------------------
// Kernel B: pooling + cnt-MLP + sigmoid + first GRU(go, h0) -> h_state[B,H]
// 64 blocks x 128 threads.
// ---------------------------------------------------------------------------
__global__ __launch_bounds__(128) void init_state_kernel(
    const float* __restrict__ reprs_ws,
    const float* __restrict__ cntW1, const float* __restrict__ cntb1,
    const float* __restrict__ cntW2, const float* __restrict__ cntb2,
    const float* __restrict__ Wih,   const float* __restrict__ Whh,
    const float* __restrict__ bih,   const float* __restrict__ bhh,
    const float* __restrict__ go,
    float* __restrict__ h_state)
{
    __shared__ float cat[2 * H_];
    __shared__ float hid[H_];
    __shared__ float h0s[H_];
    __shared__ float gos[H_];

    const int h = threadIdx.x, b = blockIdx.x;
    const float* rb = reprs_ws + (size_t)b * S_ * H_;

    float mx = -FLT_MAX, sm = 0.f;
    for (int s = 0; s < S_; ++s) {
        float v = rb[s * H_ + h];
        mx = v > mx ? v : mx;
        sm += v;
    }
    cat[h]       = mx;
    cat[H_ + h]  = sm * (1.f / (float)S_);
    gos[h]       = go[h];
    __syncthreads();

    float acc = cntb1[h];
    for (int k = 0; k < 2 * H_; ++k) acc += cat[k] * cntW1[k * H_ + h];
    hid[h] = acc > 0.f ? acc : 0.f;
    __syncthreads();

    acc = cntb2[h];
    for (int k = 0; k < H_; ++k) acc += hid[k] * cntW2[k * H_ + h];
    float h0v = 1.f / (1.f + expf(-acc));
    h0s[h] = h0v;
    __syncthreads();

    float gi[3], gh[3];
    #pragma unroll
    for (int g = 0; g < 3; ++g) {
        const float* wi = Wih + (size_t)(g * H_ + h) * H_;
        const float* wh = Whh + (size_t)(g * H_ + h) * H_;
        float ai = bih[g * H_ + h], ah = bhh[g * H_ + h];
        for (int k = 0; k < H_; ++k) { ai += gos[k] * wi[k]; ah += h0s[k] * wh[k]; }
        gi[g] = ai; gh[g] = ah;
    }
    float r = 1.f / (1.f + expf(-(gi[0] + gh[0])));
    float z = 1.f / (1.f + expf(-(gi[1] + gh[1])));
    float n = tanhf(gi[2] + r * gh[2]);
    h_state[b * H_ + h] = (1.f - z) * n + z * h0v;
}

// ---------------------------------------------------------------------------
// Kernel C: 128-step pointer decoder, one block per batch element (no grid sync).
// 64 blocks x 384 threads. Writes out[b, idx_t] = 0.9^t directly.
// ---------------------------------------------------------------------------
__global__ __launch_bounds__(384) void decode_kernel(
    const float* __restrict__ reprs_ws, const float* __restrict__ pre_ws,
    const float* __restrict__ cW1, const float* __restrict__ cW2,
    const float* __restrict__ Wih, const float* __restrict__ Whh,
    const float* __restrict__ bih, const float* __restrict__ bhh,
    const float* __restrict__ h_state, float* __restrict__ out)
{
    __shared__ float h_lds[H_], hw[H_], x_lds[H_], w2s[H_];
    __shared__ float gi_s[3 * H_], gh_s[3 * H_];
    __shared__ float sval[S_];
    __shared__ int   sidx[S_];
    __shared__ int   mask[S_];
    __shared__ int   idx_sh;

    const int t = threadIdx.x, b = blockIdx.x;
    if (t < H_) { h_lds[t] = h_state[b * H_ + t]; w2s[t] = cW2[t]; }
    if (t < S_) { mask[t] = 1; }
    __syncthreads();

    const float* preb = pre_ws + (size_t)b * S_ * H_;
    float p = 1.0f;   // 0.9^t, uniform across threads

    for (int step = 0; step < S_; ++step) {
        // hw = h @ com_W1   (com_b1 already folded into pre)
        if (t < H_) {
            float a = 0.f;
            for (int k = 0; k < H_; ++k) a += h_lds[k] * cW1[k * H_ + t];
            hw[t] = a;
        }
        __syncthreads();

        // s[j] = relu(pre[b,j,:] + hw) . com_W2   (com_b2 constant: argmax-invariant)
        if (t < S_) {
            float s;
            if (mask[t]) {
                s = 0.f;
                const float* pr = preb + t * H_;
                for (int k = 0; k < H_; ++k) {
                    float v = pr[k] + hw[k];
                    s += (v > 0.f ? v : 0.f) * w2s[k];
                }
            } else {
                s = -FLT_MAX;
            }
            sval[t] = s; sidx[t] = t;
        }
        __syncthreads();

        // argmax tree reduction, first-max (lowest index) tie-break like jnp.argmax
        for (int off = S_ / 2; off > 0; off >>= 1) {
            if (t < off) {
                float v1 = sval[t], v2 = sval[t + off];
                int   i1 = sidx[t], i2 = sidx[t + off];
                if (v2 > v1 || (v2 == v1 && i2 < i1)) { sval[t] = v2; sidx[t] = i2; }
            }
            __syncthreads();
        }
        if (t == 0) {
            int sel = sidx[0];
            idx_sh  = sel;
            mask[sel] = 0;
            out[b * S_ + sel] = p;   // 0.9 ** (selection step) == 0.9 ** argsort
        }
        __syncthreads();

        // gather selected document representation
        if (t < H_) x_lds[t] = reprs_ws[((size_t)b * S_ + idx_sh) * H_ + t];
        __syncthreads();

        // GRU gate matvecs: gi = x @ Wih^T + bih ; gh = h @ Whh^T + bhh
        if (t < 3 * H_) {
            const float* wi = Wih + (size_t)t * H_;
            const float* wh = Whh + (size_t)t * H_;
            float ai = bih[t], ah = bhh[t];
            for (int k = 0; k < H_; ++k) { ai += x_lds[k] * wi[k]; ah += h_lds[k] * wh[k]; }
            gi_s[t] = ai; gh_s[t] = ah;
        }
        __syncthreads();

        // GRU combine
        if (t < H_) {
            float r = 1.f / (1.f + expf(-(gi_s[t] + gh_s[t])));
            float z = 1.f / (1.f + expf(-(gi_s[H_ + t] + gh_s[H_ + t])));
            float n = tanhf(gi_s[2 * H_ + t] + r * gh_s[2 * H_ + t]);
            h_lds[t] = (1.f - z) * n + z * h_lds[t];
        }
        __syncthreads();

        p *= 0.9f;
    }
}

extern "C" void kernel_launch(void* const* d_in, const int* in_sizes, int n_in,
                              void* d_out, int out_size, void* d_ws, size_t ws_size,
                              hipStream_t stream) {
    const float* x       = (const float*)d_in[0];
    // d_in[1] = label  : unused by the returned output
    const float* fnn_W1  = (const float*)d_in[2];
    const float* fnn_b1  = (const float*)d_in[3];
    const float* fnn_W2  = (const float*)d_in[4];
    const float* fnn_b2  = (const float*)d_in[5];
    const float* cnt_W1  = (const float*)d_in[6];
    const float* cnt_b1  = (const float*)d_in[7];
    const float* cnt_W2  = (const float*)d_in[8];
    const float* cnt_b2  = (const float*)d_in[9];
    const float* com_W1  = (const float*)d_in[10];
    const float* com_b1  = (const float*)d_in[11];
    const float* com_W2  = (const float*)d_in[12];
    // d_in[13] = com_b2 : constant shift, argmax-invariant -> unused
    const float* gru_Wih = (const float*)d_in[14];
    const float* gru_Whh = (const float*)d_in[15];
    const float* gru_bih = (const float*)d_in[16];
    const float* gru_bhh = (const float*)d_in[17];
    const float* go      = (const float*)d_in[18];

    float* reprs_ws = (float*)d_ws;                              // [B*S, H]
    float* pre_ws   = reprs_ws + (size_t)B_ * S_ * H_;           // [B*S, H]
    float* h_state  = pre_ws   + (size_t)B_ * S_ * H_;           // [B, H]
    float* outp     = (float*)d_out;                             // [B, S]

    fused_mlp_kernel<<<(B_ * S_) / 16, 256, 0, stream>>>(
        x, fnn_W1, fnn_b1, fnn_W2, fnn_b2, com_W1, com_b1, reprs_ws, pre_ws);

    init_state_kernel<<<B_, 128, 0, stream>>>(
        reprs_ws, cnt_W1, cnt_b1, cnt_W2, cnt_b2,
        gru_Wih, gru_Whh, gru_bih, gru_bhh, go, h_state);

    decode_kernel<<<B_, 384, 0, stream>>>(
        reprs_ws, pre_ws, com_W1, com_W2,
        gru_Wih, gru_Whh, gru_bih, gru_bhh, h_state, outp);
}